// GraphSAGENet_78494822302262
// MI455X (gfx1250) — compile-verified
//
#include <hip/hip_runtime.h>

// GraphSAGE-style GNN, 5 layers, fp32 WMMA (V_WMMA_F32_16X16X4_F32) on gfx1250.
// One wave32 processes a tile of 16 nodes:
//   emb  : per node, 16x48 (gathered nbr feats) x 48x32 (We)  -> 24 WMMAs
//   agg  : tanh + max over 16 neighbor rows (8 VGPRs + shfl_xor(16))
//   vf'  : 16x64 (vin tile in LDS)            x 64x32 (Wh)    -> 32 WMMAs
// Layers are separate kernel launches; vf ping-pongs d_out <-> d_ws.

typedef __attribute__((ext_vector_type(2))) float v2f;
typedef __attribute__((ext_vector_type(8))) float v8f;

#define WMMA_F32(A, B, C) \
  __builtin_amdgcn_wmma_f32_16x16x4_f32(false, (A), false, (B), (short)0, (C), false, false)

#define NB 4
#define NN 50000
#define TILES_PER_B 3125          // 50000 / 16
#define TOTAL_TILES (NB * TILES_PER_B)
#define WPB 8                     // waves per block (256 threads)
#define NBLOCKS ((TOTAL_TILES + WPB - 1) / WPB)

__device__ __forceinline__ float fast_tanh(float x) {
  // tanh(x) = 1 - 2/(exp(2x)+1); well-behaved at +/-inf of exp
  float e = __expf(2.0f * x);
  return 1.0f - 2.0f / (e + 1.0f);
}

__global__ __launch_bounds__(256) void sage_layer(
    const float* __restrict__ vf_in,    // (B, N, 32)
    const float* __restrict__ edges,    // (B, N, 16)
    const int*   __restrict__ structure,// (N, 16, 2) int32
    const float* __restrict__ We,       // (48, 32) this layer
    const float* __restrict__ be,       // (32)
    const float* __restrict__ Wh,       // (64, 32)
    const float* __restrict__ bh,       // (32)
    float*       __restrict__ vf_out,   // (B, N, 32)
    int last) {
  __shared__ float vin[WPB][16][68];    // 16 nodes x 64 feats, padded stride

  const int lane = threadIdx.x & 31;
  const int wave = threadIdx.x >> 5;
  const int tile = blockIdx.x * WPB + wave;
  if (tile >= TOTAL_TILES) return;      // wave-uniform: EXEC stays all-ones

  const int b     = tile / TILES_PER_B;
  const int node0 = (tile - b * TILES_PER_B) * 16;
  const int col   = lane & 15;          // N-column (per 16-wide tile) / A-row
  const int half  = lane >> 4;
  const int kp    = half << 1;          // K sub-offset: 0 or 2

  // ---- Preload We B-fragments (held in VGPRs for the whole kernel) ----
  v2f bwe0[12], bwe1[12];
#pragma unroll
  for (int ks = 0; ks < 12; ++ks) {
    const int kr = ks * 4 + kp;
    bwe0[ks].x = We[kr * 32 + col];
    bwe0[ks].y = We[(kr + 1) * 32 + col];
    bwe1[ks].x = We[kr * 32 + 16 + col];
    bwe1[ks].y = We[(kr + 1) * 32 + 16 + col];
  }
  const float be0 = be[col], be1 = be[16 + col];
  const float bh0 = bh[col], bh1 = bh[16 + col];

  const long bbase = (long)b * NN;

  // ---- Phase 1: per-node neighbor embedding + max-aggregation ----
  for (int m = 0; m < 16; ++m) {
    const int n = node0 + m;
    // This lane gathers neighbor row (col) of node n
    const int vi = structure[((long)n * 16 + col) * 2 + 0];
    const int ei = structure[((long)n * 16 + col) * 2 + 1];
    const float* vrow = vf_in + (bbase + vi) * 32;
    const float* erow = edges + (bbase + ei) * 16;

    v8f c0 = {}; v8f c1 = {};
#pragma unroll
    for (int ks = 0; ks < 8; ++ks) {          // K = 0..31 from vertex feats
      v2f a = *(const v2f*)(vrow + ks * 4 + kp);
      c0 = WMMA_F32(a, bwe0[ks], c0);
      c1 = WMMA_F32(a, bwe1[ks], c1);
    }
#pragma unroll
    for (int ks = 8; ks < 12; ++ks) {         // K = 32..47 from edge feats
      v2f a = *(const v2f*)(erow + (ks - 8) * 4 + kp);
      c0 = WMMA_F32(a, bwe0[ks], c0);
      c1 = WMMA_F32(a, bwe1[ks], c1);
    }

    // tanh + bias, then max over 16 neighbor rows:
    // VGPR r, lanes<16 hold M=r; lanes>=16 hold M=r+8 (same column)
    float mx0 = -1e30f, mx1 = -1e30f;
#pragma unroll
    for (int r = 0; r < 8; ++r) {
      mx0 = fmaxf(mx0, fast_tanh(c0[r] + be0));
      mx1 = fmaxf(mx1, fast_tanh(c1[r] + be1));
    }
    mx0 = fmaxf(mx0, __shfl_xor(mx0, 16, 32));
    mx1 = fmaxf(mx1, __shfl_xor(mx1, 16, 32));

    // vin[m] = [ vf_in(node n) 0..31 | agg 0..31 ]
    if (half == 0) vin[wave][m][32 + col] = mx0;   // agg cols 0..15
    else           vin[wave][m][48 + col] = mx1;   // agg cols 16..31
    vin[wave][m][lane] = vf_in[(bbase + n) * 32 + lane];
  }

  // Make per-lane LDS writes visible to the whole wave (in-order DS + wait)
  __builtin_amdgcn_wave_barrier();
  asm volatile("s_wait_dscnt 0" ::: "memory");

  // ---- Phase 2: D(16x32) = vin(16x64) x Wh(64x32) ----
  v8f d0 = {}; v8f d1 = {};
#pragma unroll
  for (int ks = 0; ks < 16; ++ks) {
    const int kr = ks * 4 + kp;
    v2f a = *(const v2f*)(&vin[wave][col][kr]);  // 8B-aligned (kr even, pad 68)
    v2f w0, w1;
    w0.x = Wh[kr * 32 + col];       w0.y = Wh[(kr + 1) * 32 + col];
    w1.x = Wh[kr * 32 + 16 + col];  w1.y = Wh[(kr + 1) * 32 + 16 + col];
    d0 = WMMA_F32(a, w0, d0);
    d1 = WMMA_F32(a, w1, d1);
  }

  // ---- Epilogue: bias (+ tanh + row L2-normalize unless last layer) ----
#pragma unroll
  for (int r = 0; r < 8; ++r) {
    const int node = node0 + r + 8 * half;
    float t0 = d0[r] + bh0;
    float t1 = d1[r] + bh1;
    if (!last) {
      t0 = fast_tanh(t0);
      t1 = fast_tanh(t1);
      float ss = t0 * t0 + t1 * t1;        // each lane: 2 of node's 32 cols
      ss += __shfl_xor(ss, 1, 32);
      ss += __shfl_xor(ss, 2, 32);
      ss += __shfl_xor(ss, 4, 32);
      ss += __shfl_xor(ss, 8, 32);         // sum within 16-lane half = full row
      float inv = 1.0f / sqrtf(ss);
      t0 *= inv;
      t1 *= inv;
    }
    float* orow = vf_out + (bbase + node) * 32;
    orow[col]      = t0;
    orow[16 + col] = t1;
  }
}

extern "C" void kernel_launch(void* const* d_in, const int* in_sizes, int n_in,
                              void* d_out, int out_size, void* d_ws, size_t ws_size,
                              hipStream_t stream) {
  const float* vertices  = (const float*)d_in[0];   // (4, 50000, 32)
  const float* edges     = (const float*)d_in[1];   // (4, 50000, 16)
  const int*   structure = (const int*)d_in[2];     // (50000, 16, 2)
  const float* We        = (const float*)d_in[3];   // (5, 48, 32)
  const float* be        = (const float*)d_in[4];   // (5, 32)
  const float* Wh        = (const float*)d_in[5];   // (5, 64, 32)
  const float* bh        = (const float*)d_in[6];   // (5, 32)
  float* out = (float*)d_out;                       // (4, 50000, 32)
  float* ws0 = (float*)d_ws;                        // 25.6 MB scratch

  // Ping-pong so layer 4 writes d_out and never reads what it writes:
  // L0: vertices->out, L1: out->ws0, L2: ws0->out, L3: out->ws0, L4: ws0->out
  const float* in = vertices;
  float* dsts[5] = {out, ws0, out, ws0, out};
  for (int i = 0; i < 5; ++i) {
    sage_layer<<<NBLOCKS, 256, 0, stream>>>(
        in, edges, structure,
        We + (size_t)i * 48 * 32, be + (size_t)i * 32,
        Wh + (size_t)i * 64 * 32, bh + (size_t)i * 32,
        dsts[i], i == 4 ? 1 : 0);
    in = dsts[i];
  }
}